// NMS_68126771249179
// MI455X (gfx1250) — compile-verified
//
#include <hip/hip_runtime.h>
#include <stdint.h>

// NMS for (bs=4, n=4096, 6) float32 predictions [x1,y1,x2,y2,score,cls].
// One workgroup per image. Whole image lives in LDS (CDNA5: 320KB/WGP).
// Stage-in uses CDNA5 async global->LDS copies (ASYNCcnt) + s_wait_asynccnt.
// No WMMA: IoU (min/max/compare) has no matmul structure; dataset is 393KB
// total (L2-resident), so LDS residency + wave32 VALU is the optimal path.

#define NT 1024
#define BS 4
#define SCORE_THR_C 0.1f
#define IOU_THR_C 0.3f
#define MAX_COORD_C 4096.0f

__global__ __launch_bounds__(NT) void nms_gfx1250_kernel(
    const float* __restrict__ preds, float* __restrict__ out, int n)
{
    extern __shared__ unsigned char smem[];
    // Layout (bytes): preds n*24 | key n*4 | idx n*4 | bx1..by2 4*n*4 | sup n
    float*         s_pred = (float*)smem;                       // n*6
    float*         s_key  = (float*)(smem + (size_t)n * 24);    // n
    int*           s_idx  = (int*)  (smem + (size_t)n * 28);    // n
    float*         bx1    = (float*)(smem + (size_t)n * 32);
    float*         by1    = (float*)(smem + (size_t)n * 36);
    float*         bx2    = (float*)(smem + (size_t)n * 40);
    float*         by2    = (float*)(smem + (size_t)n * 44);
    unsigned char* sup    = (unsigned char*)(smem + (size_t)n * 48);

    const int tid = threadIdx.x;
    const int b   = blockIdx.x;
    const float* gimg = preds + (size_t)b * n * 6;

    // ---- Stage image into LDS with async global->LDS B128 copies (GVS mode).
    // Low 32 bits of a flat shared-aperture address are the LDS byte offset
    // (aperture id lives entirely in addr[63:32] on CDNA5).
    unsigned ldsbase = (unsigned)(uintptr_t)s_pred;
    int chunks = (n * 24) >> 4;                 // 16B per lane per instruction
    for (int c = tid; c < chunks; c += NT) {
        unsigned off      = (unsigned)c << 4;   // byte offset within image
        unsigned lds_addr = ldsbase + off;
        asm volatile("global_load_async_to_lds_b128 %0, %1, %2"
                     :: "v"(lds_addr), "v"(off), "s"(gimg) : "memory");
    }
    asm volatile("s_wait_asynccnt 0" ::: "memory");
    __syncthreads();

    // ---- Sort keys: score if valid else -inf; stable tiebreak = index asc.
    for (int i = tid; i < n; i += NT) {
        float sc = s_pred[i * 6 + 4];
        s_key[i] = (sc >= SCORE_THR_C) ? sc : -__builtin_inff();
        s_idx[i] = i;
    }
    __syncthreads();

    // ---- Bitonic sort, "ascending" under precedes := (score desc, idx asc).
    for (int k = 2; k <= n; k <<= 1) {
        for (int j = k >> 1; j > 0; j >>= 1) {
            for (int i = tid; i < n; i += NT) {
                int ixj = i ^ j;
                if (ixj > i) {
                    float ka = s_key[i], kb = s_key[ixj];
                    int   ia = s_idx[i], ib = s_idx[ixj];
                    // "a belongs after b" in final order
                    bool after = (ka < kb) || (ka == kb && ia > ib);
                    bool up = ((i & k) == 0);
                    if (up ? after : !after) {
                        s_key[i] = kb; s_key[ixj] = ka;
                        s_idx[i] = ib; s_idx[ixj] = ia;
                    }
                }
            }
            __syncthreads();
        }
    }

    // ---- Gather class-offset boxes in sorted order; clear suppression flags.
    for (int i = tid; i < n; i += NT) {
        int o = s_idx[i] * 6;
        float off = s_pred[o + 5] * MAX_COORD_C;   // cls * 4096
        bx1[i] = s_pred[o + 0] + off;
        by1[i] = s_pred[o + 1] + off;
        bx2[i] = s_pred[o + 2] + off;
        by2[i] = s_pred[o + 3] + off;
        sup[i] = 0;
    }
    __syncthreads();

    // ---- Greedy suppression. sup[i] is only ever written by iterations < i,
    // each followed by a barrier, so the keep predicate is computed uniformly
    // by all threads (same LDS words) and the barrier is uniform.
    for (int i = 0; i < n; ++i) {
        bool kp = (s_key[i] >= SCORE_THR_C) && (sup[i] == 0);
        if (kp) {
            float x1 = bx1[i], y1 = by1[i], x2 = bx2[i], y2 = by2[i];
            float ai = (x2 - x1) * (y2 - y1);
            for (int j = i + 1 + tid; j < n; j += NT) {
                float iw = fminf(x2, bx2[j]) - fmaxf(x1, bx1[j]);
                float ih = fminf(y2, by2[j]) - fmaxf(y1, by1[j]);
                float inter = fmaxf(iw, 0.0f) * fmaxf(ih, 0.0f);
                float aj = (bx2[j] - bx1[j]) * (by2[j] - by1[j]);
                // inter/(ai+aj-inter+eps) > thr, denominator > 0 always
                if (inter > IOU_THR_C * (ai + aj - inter + 1e-9f))
                    sup[j] = 1;   // j unique per thread: plain byte store
            }
            __syncthreads();
        }
    }
    __syncthreads();

    // ---- Scatter: out[orig] = pred[orig] * keep (sup[i] is final here).
    float* oimg = out + (size_t)b * n * 6;
    for (int i = tid; i < n; i += NT) {
        int o = s_idx[i] * 6;
        float m = ((s_key[i] >= SCORE_THR_C) && (sup[i] == 0)) ? 1.0f : 0.0f;
#pragma unroll
        for (int c = 0; c < 6; ++c)
            oimg[o + c] = s_pred[o + c] * m;
    }
}

extern "C" void kernel_launch(void* const* d_in, const int* in_sizes, int n_in,
                              void* d_out, int out_size, void* d_ws, size_t ws_size,
                              hipStream_t stream) {
    (void)n_in; (void)out_size; (void)d_ws; (void)ws_size;
    const float* preds = (const float*)d_in[0];
    float* out = (float*)d_out;
    int n = in_sizes[0] / (BS * 6);             // 4096
    size_t lds = (size_t)n * 48 + (size_t)n;    // arrays + sup bytes (~200KB)
    lds = (lds + 255) & ~(size_t)255;
    nms_gfx1250_kernel<<<dim3(BS), dim3(NT), lds, stream>>>(preds, out, n);
}